// Conv2D_30253749633047
// MI455X (gfx1250) — compile-verified
//
#include <hip/hip_runtime.h>

typedef float v2f __attribute__((ext_vector_type(2)));
typedef float v4f __attribute__((ext_vector_type(4)));
typedef float v8f __attribute__((ext_vector_type(8)));

#define IH 4096
#define IW 4096
#define KH 11
#define KW 11
#define OH (IH - KH + 1)   // 4086
#define OW (IW - KW + 1)   // 4086

#define KC     7                      // K chunks of 4 -> K = 28 (26 used)
#define PROWS  138                    // 8 waves * 16 rows + 10 halo
#define PCOLS  28                     // patch width in floats (16 + 11 + pad)
#define NPAIR  (KH * KC * 2 * 16)    // 2464 Toeplitz float2 fragments

// Block = 256 threads = 8 waves. Wave w computes the 16x16 output tile at
// (y = 128*blockIdx.y + 16*w, x = 16*blockIdx.x) as
//   D = sum_{kh} A_kh(16x28) * T_kh(28x16),  K chunked by 4 (f32 WMMA).
// A comes from an LDS-staged image patch; T comes from an LDS table of
// per-lane Toeplitz float2 fragments built once per block.
__global__ __launch_bounds__(256) void conv2d_wmma_f32(
    const float* __restrict__ img,
    const float* __restrict__ wgt,
    const float* __restrict__ bias,
    float* __restrict__ out) {
  __shared__ __align__(16) float patch[PROWS * PCOLS];   // 15456 B
  __shared__ __align__(8)  v2f   bpairs[NPAIR];          // 19712 B

  const int tid  = threadIdx.x;
  const int lane = tid & 31;
  const int wave = tid >> 5;
  const int mn   = lane & 15;   // A/C row index m == B/C col index n
  const int hi   = lane >> 4;   // half-wave selector (K pair)

  const int x0  = blockIdx.x * 16;    // tile col base (all waves share)
  const int y0b = blockIdx.y * 128;   // block row base
  const int y0  = y0b + wave * 16;    // this wave's tile row base

  // ---- Fill Toeplitz fragment table: pair i = ((kh*KC+kc)*2+hi)*16 + n
  //      holds (T_kh(j0,n), T_kh(j0+1,n)), j0 = 4*kc + 2*hi.
  for (int i = tid; i < NPAIR; i += 256) {
    const int n   = i & 15;
    int t         = i >> 4;
    const int phi = t & 1;  t >>= 1;
    const int kc  = t % KC;
    const int kkh = t / KC;
    const int d0  = kc * 4 + 2 * phi - n;     // = j - n
    const int d1  = d0 + 1;
    const int c0  = d0 < 0 ? 0 : (d0 > KW - 1 ? KW - 1 : d0);
    const int c1  = d1 < 0 ? 0 : (d1 > KW - 1 ? KW - 1 : d1);
    v2f p;
    p.x = (d0 >= 0 && d0 < KW) ? wgt[kkh * KW + c0] : 0.0f;
    p.y = (d1 >= 0 && d1 < KW) ? wgt[kkh * KW + c1] : 0.0f;
    bpairs[i] = p;
  }

  // ---- Stage image patch: rows y0b..y0b+137, cols x0..x0+27 (float4 tiles).
  //      Clamped OOB reads only feed outputs that the store guard discards.
  for (int i = tid; i < PROWS * (PCOLS / 4); i += 256) {
    const int r = i / (PCOLS / 4);
    const int q = i % (PCOLS / 4);
    int grow = y0b + r;         grow = grow < IH ? grow : (IH - 1);
    int gcol = x0 + 4 * q;      gcol = gcol <= (IW - 4) ? gcol : (IW - 4);
    const v4f v = *(const v4f*)(img + (long)grow * IW + gcol);
    *(v4f*)(patch + r * PCOLS + 4 * q) = v;
  }

  __syncthreads();

  // C starts at bias -> output = conv + bias.
  const float bval = bias[0];
  v8f c;
#pragma unroll
  for (int r = 0; r < 8; ++r) c[r] = bval;

  const int prow0 = wave * 16 + mn;   // this lane's A-row base in the patch

  for (int kh = 0; kh < KH; ++kh) {
    const float* arow = patch + (prow0 + kh) * PCOLS;
    const v2f*   brow = bpairs + (kh * KC * 2) * 16;

    // Load all fragments for this kh first (lets DScnt drain progressively
    // and overlap with the matrix pipe), then issue the 7 WMMAs.
    v2f a[KC], b[KC];
#pragma unroll
    for (int kc = 0; kc < KC; ++kc) {
      a[kc] = *(const v2f*)(arow + kc * 4 + 2 * hi);   // ds_load_b64
      b[kc] = brow[(kc * 2 + hi) * 16 + mn];           // ds_load_b64
    }
#pragma unroll
    for (int kc = 0; kc < KC; ++kc) {
      c = __builtin_amdgcn_wmma_f32_16x16x4_f32(
          /*neg_a=*/false, a[kc], /*neg_b=*/false, b[kc],
          /*c_mod=*/(short)0, c, /*reuse_a=*/false, /*reuse_b=*/false);
    }
  }

  // ---- Store: D element (M = r + 8*hi, N = mn) lives in c[r].
  const int xo = x0 + mn;
#pragma unroll
  for (int r = 0; r < 8; ++r) {
    const int yo = y0 + r + 8 * hi;
    if (xo < OW && yo < OH) out[(long)yo * OW + xo] = c[r];
  }
}

extern "C" void kernel_launch(void* const* d_in, const int* in_sizes, int n_in,
                              void* d_out, int out_size, void* d_ws, size_t ws_size,
                              hipStream_t stream) {
  (void)in_sizes; (void)n_in; (void)out_size; (void)d_ws; (void)ws_size;
  const float* img  = (const float*)d_in[0];
  const float* wgt  = (const float*)d_in[1];
  const float* bias = (const float*)d_in[2];
  float* out = (float*)d_out;

  dim3 grid(256, 32, 1);   // 256 x-tiles; 32 blocks * 8 waves = 256 y-tiles
  dim3 block(256, 1, 1);
  conv2d_wmma_f32<<<grid, block, 0, stream>>>(img, wgt, bias, out);
}